// MultiheadAttention_30657476559145
// MI455X (gfx1250) — compile-verified
//
#include <hip/hip_runtime.h>
#include <hip/hip_bf16.h>
#include <math.h>

// Problem constants (from reference): T=512, B=8, E=1024, H=32, HD=32
#define T_ 512
#define B_ 8
#define E_ 1024
#define H_ 32
#define HD_ 32
#define BT_ 4096  // B*T

typedef __attribute__((ext_vector_type(2))) float v2f;
typedef __attribute__((ext_vector_type(8))) float v8f;
typedef int i32x4 __attribute__((vector_size(4 * sizeof(int))));

// ---------------------------------------------------------------------------
// CDNA5 async global->LDS support (ASYNCcnt-tracked DMA-style copies).
// Probe-verified parameter types: b32 takes (int*, int*, imm, imm),
// b128 takes (int4*, int4*, imm, imm) — generic typed pointers.
// ---------------------------------------------------------------------------
#if defined(__has_builtin)
#if __has_builtin(__builtin_amdgcn_global_load_async_to_lds_b32) && \
    __has_builtin(__builtin_amdgcn_global_load_async_to_lds_b128)
#define HAVE_ASYNC 1
#endif
#endif
#ifndef HAVE_ASYNC
#define HAVE_ASYNC 0
#endif

#if HAVE_ASYNC
__device__ __forceinline__ void async_g2l_b32(const void* g, void* l) {
    __builtin_amdgcn_global_load_async_to_lds_b32((int*)g, (int*)l, 0, 0);
}
__device__ __forceinline__ void async_g2l_b128(const void* g, void* l) {
    __builtin_amdgcn_global_load_async_to_lds_b128((i32x4*)g, (i32x4*)l, 0, 0);
}
#endif

__device__ __forceinline__ void wait_async_zero() {
#if HAVE_ASYNC
#if __has_builtin(__builtin_amdgcn_s_wait_asynccnt)
    __builtin_amdgcn_s_wait_asynccnt(0);
#else
    asm volatile("s_wait_asynccnt 0" ::: "memory");
#endif
#endif
}

// D = A(16x4 f32) * B(4x16 f32) + C(16x16 f32), wave32 WMMA.
__device__ __forceinline__ v8f wmma4f32(v2f a, v2f b, v8f c) {
    // 8 args: (neg_a, A, neg_b, B, c_mod, C, reuse_a, reuse_b)
    return __builtin_amdgcn_wmma_f32_16x16x4_f32(false, a, false, b, (short)0, c,
                                                 false, false);
}

// ---------------------------------------------------------------------------
// Kernel 1: Q/K projections (tiny: 0.5 GFLOP total). One block per token.
// ---------------------------------------------------------------------------
__global__ __launch_bounds__(64) void qk_proj(
    const float* __restrict__ query,
    const float* __restrict__ Wq, const float* __restrict__ bq,
    const float* __restrict__ Wk, const float* __restrict__ bk,
    float* __restrict__ qs, float* __restrict__ kout)
{
    const int n = blockIdx.x;                 // 0..4095
    const int b = n >> 9;                     // n / T
    const int t = n & (T_ - 1);               // n % T
    const int tid = threadIdx.x;              // 0..63
    const int h = tid & (H_ - 1);

    const float4* x4 = (const float4*)(query + ((size_t)t * B_ + b) * E_);
    const float4* w4 = (const float4*)(((tid < H_) ? Wq : Wk) + (size_t)h * E_);

    float a = 0.f;
#pragma unroll 4
    for (int e = 0; e < E_ / 4; ++e) {
        float4 xv = x4[e], wv = w4[e];
        a = fmaf(xv.x, wv.x, a);
        a = fmaf(xv.y, wv.y, a);
        a = fmaf(xv.z, wv.z, a);
        a = fmaf(xv.w, wv.w, a);
    }
    if (tid < H_)
        qs[(size_t)n * H_ + h] = (a + bq[h]) * 0.17677669529663689f;  // 1/sqrt(32)
    else
        kout[(size_t)n * H_ + h] = a + bk[h];
}

// ---------------------------------------------------------------------------
// Kernel 2/4: Y[n,m] = sum_k Xrow(n)[k] * W[m,k] + bias[m]    (K = E_ = 1024)
//   IN_MODE  0: X linear [BT,E]         1: X = query [T,B,E] (gather n=b*T+t)
//   OUT_MODE 0: Y linear [BT,E]         1: Y is [T,B,E] (scatter n=b*T+t)
// Tile: 16 rows x 64 cols, 4 waves, K-chunks of 64, double-buffered async
// staging (global_load_async_to_lds_b128 when available).
// ---------------------------------------------------------------------------
#define XS_STRIDE 68   // 272 B rows -> 16-B aligned quads, conflict-free reads
#define WS_STRIDE 68

template <int IN_MODE>
__device__ __forceinline__ void gemm_stage(
    const float* __restrict__ X, const float* __restrict__ W,
    int rowBase, int colBase, int kc, float* xs, float* ws, int tid)
{
#if HAVE_ASYNC
    // X tile: 16x64 floats = 256 16-B quads (2 per thread)
    for (int it = tid; it < 16 * 16; it += 128) {
        const int r = it >> 4, q = (it & 15) << 2;
        const int n = rowBase + r;
        const float* src;
        if (IN_MODE == 1) {
            const int bb = n >> 9, tt = n & (T_ - 1);
            src = X + ((size_t)tt * B_ + bb) * E_ + kc + q;
        } else {
            src = X + (size_t)n * E_ + kc + q;
        }
        async_g2l_b128(src, &xs[r * XS_STRIDE + q]);
    }
    // W tile: 64x64 floats = 1024 quads (8 per thread)
    for (int it = tid; it < 64 * 16; it += 128) {
        const int c = it >> 4, q = (it & 15) << 2;
        async_g2l_b128(W + (size_t)(colBase + c) * E_ + kc + q,
                       &ws[c * WS_STRIDE + q]);
    }
#else
    for (int it = tid; it < 16 * 64; it += 128) {
        const int r = it >> 6, k = it & 63;
        const int n = rowBase + r;
        const float* src;
        if (IN_MODE == 1) {
            const int bb = n >> 9, tt = n & (T_ - 1);
            src = X + ((size_t)tt * B_ + bb) * E_ + kc + k;
        } else {
            src = X + (size_t)n * E_ + kc + k;
        }
        xs[r * XS_STRIDE + k] = *src;
    }
    for (int it = tid; it < 64 * 64; it += 128) {
        const int c = it >> 6, k = it & 63;
        ws[c * WS_STRIDE + k] = W[(size_t)(colBase + c) * E_ + kc + k];
    }
#endif
}

template <int IN_MODE, int OUT_MODE>
__global__ __launch_bounds__(128) void gemm_nt(
    const float* __restrict__ X, const float* __restrict__ W,
    const float* __restrict__ bias, float* __restrict__ Y)
{
    __shared__ __align__(16) float Xs[2][16 * XS_STRIDE];
    __shared__ __align__(16) float Ws[2][64 * WS_STRIDE];

    const int rowBase = blockIdx.x * 16;
    const int colBase = blockIdx.y * 64;
    const int tid = threadIdx.x;              // 0..127
    const int l  = tid & 31;
    const int wv = tid >> 5;                  // wave 0..3 -> 16-col subtile
    const int lm = l & 15;
    const int lh = l >> 4;                    // lane half

    v8f acc = {};

    // Prologue: stage chunk 0
    gemm_stage<IN_MODE>(X, W, rowBase, colBase, 0, Xs[0], Ws[0], tid);
    wait_async_zero();
    __syncthreads();

    for (int kc = 0; kc < E_; kc += 64) {
        const int cur = (kc >> 6) & 1;
        // Issue next chunk's staging early; overlap with current compute.
        if (kc + 64 < E_)
            gemm_stage<IN_MODE>(X, W, rowBase, colBase, kc + 64,
                                Xs[cur ^ 1], Ws[cur ^ 1], tid);

        const float* xb = &Xs[cur][lm * XS_STRIDE + 2 * lh];
        const float* wb = &Ws[cur][(wv * 16 + lm) * WS_STRIDE + 2 * lh];
#pragma unroll
        for (int ks = 0; ks < 64; ks += 4) {
            v2f a, bv2;
            a.x = xb[ks];     a.y = xb[ks + 1];
            bv2.x = wb[ks];   bv2.y = wb[ks + 1];
            acc = wmma4f32(a, bv2, acc);
        }
        wait_async_zero();   // next chunk landed (no-op on last iteration)
        __syncthreads();
    }

    // Epilogue: bias + store. C layout: row = r + 8*lh, col = wv*16 + lm
    const int m = colBase + wv * 16 + lm;
    const float bm = bias[m];
#pragma unroll
    for (int r = 0; r < 8; ++r) {
        const int n = rowBase + r + 8 * lh;
        float val = acc[r] + bm;
        if (OUT_MODE == 1) {
            const int bb = n >> 9, tt = n & (T_ - 1);
            Y[((size_t)tt * B_ + bb) * E_ + m] = val;
        } else {
            Y[(size_t)n * E_ + m] = val;
        }
    }
}

// ---------------------------------------------------------------------------
// Kernel 3: fused talking-heads attention.
// One block = (b, 16-row i-tile). 8 waves. Streams j in 16-tiles with
// double-buffered bias staging (the 268 MB attn_bias stream is the HBM
// bottleneck -> async copy of tile t+1 overlaps all compute of tile t).
// Per j-tile:
//   PA: score WMMAs  S[h',i,j] = bias + sum_h k[j,h]*(Wt[h',h]*q[i,h])
//       (bias tile is pre-staged in S and used as the WMMA C-init)
//   PB: online softmax per (h,i): rescale factor F, S := exp(S - m_new)
//   PC: per-head P·V WMMAs into register accumulators (rescaled by F)
// ---------------------------------------------------------------------------
#define S_H 289   // head stride in S   (17*17, odd -> conflict-free)
#define S_I 17    // i stride in S
#define K33 33
#define F17 17

__device__ __forceinline__ void stage_bias(
    const float* __restrict__ bias_base, int j0, float* sbuf, int tid)
{
    for (int it = tid; it < H_ * 16 * 16; it += 256) {
        const int h = it >> 8, i = (it >> 4) & 15, j = it & 15;
        const float* g = bias_base + ((size_t)h * T_ + i) * T_ + j0 + j;
#if HAVE_ASYNC
        async_g2l_b32(g, &sbuf[h * S_H + i * S_I + j]);
#else
        sbuf[h * S_H + i * S_I + j] = *g;
#endif
    }
}

__global__ __launch_bounds__(256) void attn_fused(
    const float* __restrict__ qs, const float* __restrict__ kmat,
    const float* __restrict__ v, const float* __restrict__ Wt,
    const float* __restrict__ attn_bias, float* __restrict__ ctx)
{
    __shared__ __align__(16) float S[2][H_ * S_H];   // bias -> scores -> probs
    __shared__ float kld[2][16 * K33];               // k tile [j][h]
    __shared__ float qld[16 * K33];                  // q tile [i][h]
    __shared__ float wt[H_ * K33];                   // Wt     [h'][h]
    __shared__ float Fb[H_ * F17];                   // rescale per (h,i)
    __shared__ float Mst[H_ * F17];                  // running max
    __shared__ float Sst[H_ * F17];                  // running sum

    const int i0 = blockIdx.x * 16;     // i-tile base
    const int b  = blockIdx.y;          // batch
    const int tid = threadIdx.x;        // 0..255
    const int l  = tid & 31;
    const int wv = tid >> 5;            // wave 0..7
    const int lm = l & 15;
    const int lh = l >> 4;

    const float* bias_base = attn_bias + (((size_t)b * H_) * T_ + i0) * T_;

    // ---- block-wide init ----
    for (int it = tid; it < 16 * H_; it += 256) {
        const int i = it >> 5, h = it & (H_ - 1);
        qld[i * K33 + h] = qs[((size_t)b * T_ + i0 + i) * H_ + h];
    }
    for (int it = tid; it < H_ * H_; it += 256) {
        const int hp = it >> 5, h = it & (H_ - 1);
        wt[hp * K33 + h] = Wt[hp * H_ + h];
    }
    for (int it = tid; it < H_ * 16; it += 256) {
        const int h = it >> 4, i = it & 15;
        Mst[h * F17 + i] = -INFINITY;
        Sst[h * F17 + i] = 0.f;
    }

    v8f acc[4][2];                       // 4 heads per wave x 2 d-subtiles
#pragma unroll
    for (int hh = 0; hh < 4; ++hh)
#pragma unroll
        for (int dt = 0; dt < 2; ++dt) acc[hh][dt] = (v8f){};

    // Prologue: stage tile 0
    stage_bias(bias_base, 0, S[0], tid);
    for (int it = tid; it < 16 * H_; it += 256) {
        const int j = it >> 5, h = it & (H_ - 1);
        kld[0][j * K33 + h] = kmat[((size_t)b * T_ + j) * H_ + h];
    }
    wait_async_zero();
    __syncthreads();

    for (int jt = 0; jt < T_ / 16; ++jt) {
        const int j0 = jt * 16;
        const int cur = jt & 1;

        // ---- issue next tile's staging (overlaps PA/PB/PC below) ----
        if (jt + 1 < T_ / 16) {
            stage_bias(bias_base, j0 + 16, S[cur ^ 1], tid);
            for (int it = tid; it < 16 * H_; it += 256) {
                const int j = it >> 5, h = it & (H_ - 1);
                kld[cur ^ 1][j * K33 + h] =
                    kmat[((size_t)b * T_ + j0 + 16 + j) * H_ + h];
            }
        }

        // ---- PA: scores (2 i-rows per wave, 2 h'-subtiles, K=32 heads) ----
        float* Sc = S[cur];
        const float* kc = kld[cur];
#pragma unroll
        for (int ii = 0; ii < 2; ++ii) {
            const int i = wv * 2 + ii;
#pragma unroll
            for (int hp0 = 0; hp0 < 32; hp0 += 16) {
                float* Srow = &Sc[(hp0 + lm) * S_H + i * S_I];
                v8f c;
#pragma unroll
                for (int r = 0; r < 8; ++r) c[r] = Srow[r + 8 * lh];  // bias C-init
#pragma unroll
                for (int k0 = 0; k0 < H_; k0 += 4) {
                    const int k = k0 + 2 * lh;
                    v2f a, bb;
                    a.x = kc[lm * K33 + k];
                    a.y = kc[lm * K33 + k + 1];
                    bb.x = wt[(hp0 + lm) * K33 + k]     * qld[i * K33 + k];
                    bb.y = wt[(hp0 + lm) * K33 + k + 1] * qld[i * K33 + k + 1];
                    c = wmma4f32(a, bb, c);
                }
#pragma unroll
                for (int r = 0; r < 8; ++r) Srow[r + 8 * lh] = c[r];
            }
        }
        __syncthreads();

        // ---- PB: online softmax over this j-tile; S := exp(S - m_new) ----
        for (int e = tid; e < H_ * 16; e += 256) {
            const int h = e >> 4, i = e & 15;
            float* Sr = &Sc[h * S_H + i * S_I];
            float mloc = Sr[0];
#pragma unroll
            for (int j = 1; j < 16; ++j) mloc = fmaxf(mloc, Sr[j]);
            const float mold = Mst[h * F17 + i];
            const float mnew = fmaxf(mold, mloc);
            const float f = __expf(mold - mnew);   // 0 on first tile
            float s = Sst[h * F17 + i] * f;
#pragma unroll
            for (int j = 0; j < 16; ++j) {
                const float p = __expf(Sr[j] - mnew);
                Sr[j] = p;
                s += p;
            }
            Mst[h * F17 + i] = mnew;
            Sst[h * F17 + i] = s;
            Fb[h * F17 + i]  = f;
        }
        __syncthreads();

        // ---- PC: out[i,d] += P[i,j] * V[j,d] per head (4 heads/wave) ----
#pragma unroll
        for (int hh = 0; hh < 4; ++hh) {
            const int h = wv * 4 + hh;
            float fr[8];
#pragma unroll
            for (int r = 0; r < 8; ++r) fr[r] = Fb[h * F17 + r + 8 * lh];
            const float* Sh = &Sc[h * S_H];
#pragma unroll
            for (int dt = 0; dt < 2; ++dt) {
                v8f c = acc[hh][dt];
#pragma unroll
                for (int r = 0; r < 8; ++r) c[r] *= fr[r];
#pragma unroll
                for (int kk = 0; kk < 16; kk += 4) {
                    const int jl = kk + 2 * lh;
                    v2f a, bb;
                    a.x = Sh[lm * S_I + jl];
                    a.y = Sh[lm * S_I + jl + 1];
                    const float* vp =
                        &v[(((size_t)b * T_ + j0 + jl) * H_ + h) * HD_ + dt * 16 + lm];
                    bb.x = vp[0];
                    bb.y = vp[H_ * HD_];     // next j row
                    c = wmma4f32(a, bb, c);
                }
                acc[hh][dt] = c;
            }
        }

        wait_async_zero();   // tile jt+1 fully in LDS (issued at loop top)
        __syncthreads();
    }

    // ---- epilogue: normalize and store context ----
#pragma unroll
    for (int hh = 0; hh < 4; ++hh) {
        const int h = wv * 4 + hh;
        float sr[8];
#pragma unroll
        for (int r = 0; r < 8; ++r) sr[r] = 1.f / Sst[h * F17 + r + 8 * lh];
#pragma unroll
        for (int dt = 0; dt < 2; ++dt) {
            v8f c = acc[hh][dt];
#pragma unroll
            for (int r = 0; r < 8; ++r) {
                const int i = i0 + r + 8 * lh;
                ctx[((size_t)b * T_ + i) * E_ + h * HD_ + dt * 16 + lm] = c[r] * sr[r];
            }
        }
    }
}

// ---------------------------------------------------------------------------
// Launch
// ---------------------------------------------------------------------------
extern "C" void kernel_launch(void* const* d_in, const int* in_sizes, int n_in,
                              void* d_out, int out_size, void* d_ws, size_t ws_size,
                              hipStream_t stream) {
    const float* query     = (const float*)d_in[0];
    const float* attn_bias = (const float*)d_in[1];
    const float* Wq = (const float*)d_in[2];
    const float* bq = (const float*)d_in[3];
    const float* Wk = (const float*)d_in[4];
    const float* bk = (const float*)d_in[5];
    const float* Wv = (const float*)d_in[6];
    const float* bv = (const float*)d_in[7];
    const float* Wt = (const float*)d_in[8];
    const float* Wo = (const float*)d_in[9];
    const float* bo = (const float*)d_in[10];

    float* ws  = (float*)d_ws;
    float* qs  = ws;                                   // B*T*H   = 131072
    float* kk  = ws + (size_t)BT_ * H_;                // B*T*H   = 131072
    float* vv  = kk + (size_t)BT_ * H_;                // B*T*E   = 4194304
    float* ctx = vv + (size_t)BT_ * E_;                // B*T*E   = 4194304

    // 1) Q/K projections
    qk_proj<<<BT_, 64, 0, stream>>>(query, Wq, bq, Wk, bk, qs, kk);

    // 2) V projection:  vv[n, e'] = x[n,:]·Wv[e',:] + bv
    gemm_nt<1, 0><<<dim3(BT_ / 16, E_ / 64), 128, 0, stream>>>(query, Wv, bv, vv);

    // 3) fused talking-heads attention -> ctx [B,T,E]
    attn_fused<<<dim3(T_ / 16, B_), 256, 0, stream>>>(qs, kk, vv, Wt, attn_bias, ctx);

    // 4) output projection, scattered to [T,B,E]
    gemm_nt<0, 1><<<dim3(BT_ / 16, E_ / 64), 128, 0, stream>>>(ctx, Wo, bo,
                                                               (float*)d_out);
}